// VectorQuantization_21758304321728
// MI455X (gfx1250) — compile-verified
//
#include <hip/hip_runtime.h>
#include <hip/hip_bf16.h>

// Vector quantization argmin: for each of N tokens find nearest of K codebook
// rows (D=128). argmin ||x-c||^2 == argmin (c^2 - 2 x.c); the x.c GEMM runs on
// V_WMMA_F32_16X16X4_F32 (exact fp32), argmin fused in the epilogue.
//
// Round 3: explicit 4-chain lockstep over k-step pairs with a double-buffered
// fragment group (4 ds_load_2addr_b64 in flight across each 8-WMMA group) to
// deepen the LDS prefetch distance beyond the compiler's dscnt<=1 schedule.
// Codebook tiles stay double-buffered in LDS via GLOBAL_LOAD_ASYNC_TO_LDS_B128.

#define NTOK  32768
#define DIM   128
#define NCODE 4096

#define COLTILE 64      // codebook rows staged per LDS tile
#define LDS_STRIDE 132  // 128 + 4 pad -> bank step 4 -> conflict-free b64 reads

typedef __attribute__((ext_vector_type(2))) float v2f;
typedef __attribute__((ext_vector_type(8))) float v8f;

// ---------------------------------------------------------------- helpers
__device__ __forceinline__ void vq_wait_async0()
{
#if defined(__has_builtin) && __has_builtin(__builtin_amdgcn_s_wait_asynccnt)
    __builtin_amdgcn_s_wait_asynccnt(0);
#else
    asm volatile("s_wait_asynccnt 0x0" ::: "memory");
#endif
}

// 16B async copy global -> LDS, per-lane addresses (GV form), ASYNCcnt-tracked.
__device__ __forceinline__ void vq_async_copy_b128(unsigned lds_off, const void* gaddr)
{
    asm volatile("global_load_async_to_lds_b128 %0, %1, off"
                 :: "v"(lds_off), "v"(gaddr)
                 : "memory");
}

// ---------------------------------------------------------------- c^2 kernel
__global__ __launch_bounds__(256)
void vq_c2_kernel(const float* __restrict__ cb, float* __restrict__ c2)
{
    int k = blockIdx.x * blockDim.x + threadIdx.x;
    if (k >= NCODE) return;
    const float4* p = (const float4*)(cb + (size_t)k * DIM);
    float s = 0.f;
#pragma unroll
    for (int i = 0; i < DIM / 4; ++i) {
        float4 v = p[i];
        s += v.x * v.x + v.y * v.y + v.z * v.z + v.w * v.w;
    }
    c2[k] = s;
}

// ------------------------------------------------------------- main kernel
// Block: 256 threads = 8 waves. Wave w owns token rows [blk*128 + w*16, +16).
// Sweeps all 4096 codebook columns in 64-wide LDS tiles shared by all waves.
__global__ __launch_bounds__(256)
void vq_argmin_kernel(const float* __restrict__ x,
                      const float* __restrict__ cb,
                      const float* __restrict__ c2,
                      int* __restrict__ out)
{
    __shared__ float lb[2][COLTILE * LDS_STRIDE];   // 2 x 33,792 B (double buffer)

    const int tid  = threadIdx.x;
    const int wave = tid >> 5;
    const int lane = tid & 31;
    const int l16  = lane & 15;
    const int half = lane >> 4;           // 0: rows 0-7 of tile, 1: rows 8-15
    const int rowBase = blockIdx.x * 128 + wave * 16;

    // ---- A fragments: 16 rows x 128 dims, ISA 16x4 f32 layout.
    // lane<16 : VGPR0=A[m][k], VGPR1=A[m][k+1]; lane>=16: k+2, k+3.
    v2f afrag[32];
    {
        const float* xrow = x + (size_t)(rowBase + l16) * DIM + half * 2;
#pragma unroll
        for (int t = 0; t < 32; ++t)
            afrag[t] = *(const v2f*)(xrow + 4 * t);   // global_load_b64
    }

    float bestV[8];
    int   bestI[8];
#pragma unroll
    for (int i = 0; i < 8; ++i) { bestV[i] = 3.402823466e38f; bestI[i] = 0x7fffffff; }

    // ---- tile staging: 64 rows x 128 floats; 256 thr x 8 x async b128,
    //      per-lane LDS addresses preserve the padded (conflict-free) stride.
    auto stage_tile = [&](int buf, int c0) {
#pragma unroll
        for (int i = 0; i < 8; ++i) {
            int f = tid * 4 + i * 1024;           // 8192 floats total
            int r = f >> 7;                       // tile row (codebook row)
            int d = f & 127;                      // dim
            const float* g = cb + (size_t)(c0 + r) * DIM + d;
            unsigned ldsoff = (unsigned)(uintptr_t)(&lb[buf][r * LDS_STRIDE + d]);
            vq_async_copy_b128(ldsoff, (const void*)g);
        }
    };

    // prologue: stage first tile, wait, barrier
    stage_tile(0, 0);
    vq_wait_async0();
    __syncthreads();

    int cur = 0;
    for (int c0 = 0; c0 < NCODE; c0 += COLTILE) {
        // kick async copy of the NEXT tile into the other buffer (overlapped)
        if (c0 + COLTILE < NCODE)
            stage_tile(cur ^ 1, c0 + COLTILE);

        const float* lbuf = &lb[cur][0];

        // 4 column chains (16 cols each) processed in lockstep over k.
        const float* bb[4];
        float c2v[4];
#pragma unroll
        for (int c = 0; c < 4; ++c) {
            bb[c]  = &lbuf[(c * 16 + l16) * LDS_STRIDE + half * 2];
            c2v[c] = c2[c0 + c * 16 + l16];
        }

        v8f acc[4];
#pragma unroll
        for (int c = 0; c < 4; ++c)
            acc[c] = v8f{0.f, 0.f, 0.f, 0.f, 0.f, 0.f, 0.f, 0.f};

        // Fragment groups: group t2 holds k-steps {2*t2, 2*t2+1} for all 4
        // chains (8 v2f = 4 ds_load_2addr_b64). Double-buffered so the next
        // group's 4 loads are in flight across the current 8-WMMA group.
        v2f bf[2][4][2];
#pragma unroll
        for (int c = 0; c < 4; ++c) {
            bf[0][c][0] = *(const v2f*)(bb[c] + 0);
            bf[0][c][1] = *(const v2f*)(bb[c] + 4);
        }

#pragma unroll
        for (int t2 = 0; t2 < 16; ++t2) {
            const int curb = t2 & 1, nxtb = curb ^ 1;
            if (t2 < 15) {
#pragma unroll
                for (int c = 0; c < 4; ++c) {
                    bf[nxtb][c][0] = *(const v2f*)(bb[c] + (t2 + 1) * 8);
                    bf[nxtb][c][1] = *(const v2f*)(bb[c] + (t2 + 1) * 8 + 4);
                }
            }
#pragma unroll
            for (int c = 0; c < 4; ++c) {
                acc[c] = __builtin_amdgcn_wmma_f32_16x16x4_f32(
                    false, afrag[2 * t2],     false, bf[curb][c][0],
                    (short)0, acc[c], false, false);
                acc[c] = __builtin_amdgcn_wmma_f32_16x16x4_f32(
                    false, afrag[2 * t2 + 1], false, bf[curb][c][1],
                    (short)0, acc[c], false, false);
            }
        }

        // epilogue: key = c^2 - 2*dot ; per-lane running argmin.
        // C/D layout: VGPR i, lane L -> row (i + 8*half), col (L%16).
        // Chains visited in increasing column order; strict '<' keeps the
        // earliest (lowest) index on ties, like jnp.argmin.
#pragma unroll
        for (int c = 0; c < 4; ++c) {
            const int col = c0 + c * 16 + l16;
#pragma unroll
            for (int i = 0; i < 8; ++i) {
                float key = __builtin_fmaf(-2.0f, acc[c][i], c2v[c]);
                if (key < bestV[i]) { bestV[i] = key; bestI[i] = col; }
            }
        }

        // next tile's async writes must land; all waves done reading cur.
        vq_wait_async0();
        __syncthreads();
        cur ^= 1;
    }

    // ---- cross-lane argmin within each 16-lane half (masks 8..1 never cross
    // the half boundary); tie-break toward lower index like jnp.argmin.
#pragma unroll
    for (int i = 0; i < 8; ++i) {
        float v = bestV[i];
        int   idx = bestI[i];
#pragma unroll
        for (int m = 8; m >= 1; m >>= 1) {
            float ov = __shfl_xor(v, m, 32);
            int   oi = __shfl_xor(idx, m, 32);
            if (ov < v || (ov == v && oi < idx)) { v = ov; idx = oi; }
        }
        if (l16 == 0)
            out[rowBase + half * 8 + i] = idx;
    }
}

// ------------------------------------------------------------------ launch
extern "C" void kernel_launch(void* const* d_in, const int* in_sizes, int n_in,
                              void* d_out, int out_size, void* d_ws, size_t ws_size,
                              hipStream_t stream)
{
    const float* x  = (const float*)d_in[0];   // [N, D] fp32
    const float* cb = (const float*)d_in[1];   // [K, D] fp32
    float* c2 = (float*)d_ws;                  // K floats of scratch (16 KB)
    int*   out = (int*)d_out;                  // N int32 indices

    vq_c2_kernel<<<NCODE / 256, 256, 0, stream>>>(cb, c2);
    vq_argmin_kernel<<<NTOK / 128, 256, 0, stream>>>(x, cb, c2, out);
}